// LF_62362925138441
// MI455X (gfx1250) — compile-verified
//
#include <hip/hip_runtime.h>

// GINE-style layer for MI455X (gfx1250, wave32).
// All GEMMs use V_WMMA_F32_16X16X4_F32 (fp32 A/B, fp32 accum) to match the
// fp32 reference numerics. Workload is L2/memory bound (~10 GFLOP, ~112 MB
// mandatory HBM traffic; h0/h1/agg intermediates stay in 192 MB L2), so f32
// WMMA throughput is ample and bf16 down-conversion would only lose accuracy.

typedef __attribute__((ext_vector_type(2))) float v2f;
typedef __attribute__((ext_vector_type(8))) float v8f;

#define DD 128   // node feature dim
#define DAA 16   // edge-attr dim

__device__ __forceinline__ v2f ld2(const float* p) { return *(const v2f*)p; }

// ---------------------------------------------------------------------------
// out[16-row tile, 16-col slab] = op(A) @ W + bias.
// FUSED_RELU=false: out = X @ W + bias
// FUSED_RELU=true : out = relu((X*(1+eps[0]) + X2) @ W + bias)
// One wave owns one 16-wide column slab; the 32 B-fragments (K=128 in steps
// of 4) are register-resident (64 VGPRs) and reused across row tiles via a
// grid-stride loop, so W is read from global exactly once per wave.
// A-fragment layout (ISA 7.12.2, 32-bit A 16x4): lanes 0-15 hold K={4k,4k+1},
// lanes 16-31 hold K={4k+2,4k+3}; row = lane&15.
// C/D layout: VGPR r, lanes 0-15 -> M=r, lanes 16-31 -> M=r+8.
// ---------------------------------------------------------------------------
template <bool FUSED_RELU>
__global__ __launch_bounds__(256) void gemm16_f32wmma_kernel(
    const float* __restrict__ X, const float* __restrict__ X2,
    const float* __restrict__ epsP,
    const float* __restrict__ W, const float* __restrict__ bias,
    float* __restrict__ Out, int numRowTiles)
{
  const int lane = threadIdx.x & 31;
  const int wave = threadIdx.x >> 5;
  const int half = lane >> 4;   // K sub-select for A/B fragments
  const int l16  = lane & 15;

  const int slab = blockIdx.x & 7;        // 8 column slabs of 16 (D=128)
  const int rgrp = blockIdx.x >> 3;
  const int waveStride = (gridDim.x >> 3) * 8;
  const int col = slab * 16 + l16;

  float scale = 1.0f;
  if (FUSED_RELU) scale = 1.0f + epsP[0];

  // Register-resident B fragments: B tile is 4x16 (KxN); lane half selects
  // K rows {4k,4k+1} vs {4k+2,4k+3}, striped over lanes by N.
  v2f bw[32];
#pragma unroll
  for (int kt = 0; kt < 32; ++kt) {
    const float* wp = W + (kt * 4 + 2 * half) * DD + col;
    bw[kt].x = wp[0];
    bw[kt].y = wp[DD];
  }
  const float bv = bias[col];

  for (int tile = rgrp * 8 + wave; tile < numRowTiles; tile += waveStride) {
    // Single 64-bit base per tile; all inner accesses use constant offsets.
    const float* ap  = X + (size_t)(tile * 16 + l16) * DD + 2 * half;
    const float* ap2 = FUSED_RELU
                         ? X2 + (size_t)(tile * 16 + l16) * DD + 2 * half
                         : nullptr;
    v8f acc = {};
#pragma unroll
    for (int kt = 0; kt < 32; ++kt) {
      v2f af = ld2(ap + kt * 4);
      if (FUSED_RELU) {
        v2f ag = ld2(ap2 + kt * 4);
        af.x = fmaf(af.x, scale, ag.x);   // x*(1+eps) + agg
        af.y = fmaf(af.y, scale, ag.y);
      }
      acc = __builtin_amdgcn_wmma_f32_16x16x4_f32(
          /*neg_a=*/false, af, /*neg_b=*/false, bw[kt],
          /*c_mod=*/(short)0, acc, /*reuse_a=*/false, /*reuse_b=*/false);
    }
    float* op = Out + (size_t)(tile * 16 + 8 * half) * DD + col;
#pragma unroll
    for (int r = 0; r < 8; ++r) {        // constant byte offsets r*512
      float v = acc[r] + bv;
      if (FUSED_RELU) v = fmaxf(v, 0.0f);
      op[r * DD] = v;
    }
  }
}

// ---------------------------------------------------------------------------
// Edge kernel: per wave, one tile of 16 edges.
//   emb  = a[e0:e0+16, :] @ Wa + ba            (WMMA, K=16 -> 4 steps)
//   msg  = relu(h0[src0] + h1[src1] + emb)     (gather, L2-resident)
//   agg[dst] += msg                            (global_atomic_add_f32)
// Wa fragments (4 ksteps x 8 ntiles, 64 VGPRs) are register-resident.
// Gather/atomic addressing: one 64-bit base per edge row, constant 64-byte
// column offsets (folds into the instruction's 24-bit immediate).
// ---------------------------------------------------------------------------
__global__ __launch_bounds__(256) void edge_msg_kernel(
    const float* __restrict__ h0, const float* __restrict__ h1,
    const float* __restrict__ a,  const float* __restrict__ Wa,
    const float* __restrict__ ba, const int* __restrict__ index,
    float* __restrict__ agg, int numTiles, int E)
{
  const int lane = threadIdx.x & 31;
  const int wave = threadIdx.x >> 5;
  const int half = lane >> 4;
  const int l16  = lane & 15;

  v2f bw[4][8];
#pragma unroll
  for (int kt = 0; kt < 4; ++kt)
#pragma unroll
    for (int nt = 0; nt < 8; ++nt) {
      const float* wp = Wa + (kt * 4 + 2 * half) * DD + nt * 16 + l16;
      bw[kt][nt].x = wp[0];
      bw[kt][nt].y = wp[DD];
    }
  float bb[8];
#pragma unroll
  for (int nt = 0; nt < 8; ++nt) bb[nt] = ba[nt * 16 + l16];

  const int waveGlobal = blockIdx.x * 8 + wave;
  const int nWaves = gridDim.x * 8;

  for (int t = waveGlobal; t < numTiles; t += nWaves) {
    const int e0 = t * 16;
    v8f acc[8] = {};
    const float* ap = a + (size_t)(e0 + l16) * DAA + 2 * half;
#pragma unroll
    for (int kt = 0; kt < 4; ++kt) {
      v2f af = ld2(ap + kt * 4);
#pragma unroll
      for (int nt = 0; nt < 8; ++nt) {
        acc[nt] = __builtin_amdgcn_wmma_f32_16x16x4_f32(
            false, af, false, bw[kt][nt], (short)0, acc[nt], false, false);
      }
    }
#pragma unroll
    for (int r = 0; r < 8; ++r) {
      const int e  = e0 + r + 8 * half;           // edge for this acc row
      const int d  = index[e];                    // dst
      const int s0 = index[E + e];                // src0
      const int s1 = index[2 * E + e];            // src1
      const float* p0 = h0 + (size_t)s0 * DD + l16;
      const float* p1 = h1 + (size_t)s1 * DD + l16;
      float* pd = agg + (size_t)d * DD + l16;
#pragma unroll
      for (int nt = 0; nt < 8; ++nt) {            // constant byte offsets nt*64
        float v = acc[nt][r] + bb[nt] + p0[nt * 16] + p1[nt * 16];
        v = fmaxf(v, 0.0f);
        atomicAdd(pd + nt * 16, v);
      }
    }
  }
}

// ---------------------------------------------------------------------------
extern "C" void kernel_launch(void* const* d_in, const int* in_sizes, int n_in,
                              void* d_out, int out_size, void* d_ws, size_t ws_size,
                              hipStream_t stream) {
  const float* x     = (const float*)d_in[0];
  const int*   index = (const int*)d_in[1];
  const float* a     = (const float*)d_in[2];
  const float* W0    = (const float*)d_in[3];
  const float* b0    = (const float*)d_in[4];
  const float* W1    = (const float*)d_in[5];
  const float* b1    = (const float*)d_in[6];
  const float* Wa    = (const float*)d_in[7];
  const float* ba    = (const float*)d_in[8];
  const float* eps   = (const float*)d_in[9];
  const float* W_in  = (const float*)d_in[10];
  const float* b_in  = (const float*)d_in[11];
  const float* W_out = (const float*)d_in[12];
  const float* b_out = (const float*)d_in[13];
  float* out = (float*)d_out;

  const int N = in_sizes[0] / DD;   // 50000 (multiple of 16)
  const int E = in_sizes[1] / 3;    // 800000 (multiple of 16)
  const int rowTiles  = N / 16;
  const int edgeTiles = E / 16;

  // Workspace: h0 | h1 | agg  (25.6 MB each; all L2-resident on MI455X)
  float* h0  = (float*)d_ws;
  float* h1  = h0 + (size_t)N * DD;
  float* agg = h1 + (size_t)N * DD;
  float* tmp = h0;  // h0 region reused for MLP intermediate after edge pass

  // agg must be zeroed every call (deterministic, graph-capturable)
  hipMemsetAsync(agg, 0, (size_t)N * DD * sizeof(float), stream);

  dim3 blk(256);
  dim3 gGemm(48 * 8);   // 8 col slabs x 48 row groups -> 384 waves/slab
  dim3 gEdge(256);      // 2048 waves, ~24 edge tiles each

  gemm16_f32wmma_kernel<false><<<gGemm, blk, 0, stream>>>(
      x, nullptr, nullptr, W0, b0, h0, rowTiles);
  gemm16_f32wmma_kernel<false><<<gGemm, blk, 0, stream>>>(
      x, nullptr, nullptr, W1, b1, h1, rowTiles);
  edge_msg_kernel<<<gEdge, blk, 0, stream>>>(h0, h1, a, Wa, ba, index, agg,
                                             edgeTiles, E);
  gemm16_f32wmma_kernel<true><<<gGemm, blk, 0, stream>>>(
      x, agg, eps, W_in, b_in, tmp, rowTiles);
  gemm16_f32wmma_kernel<false><<<gGemm, blk, 0, stream>>>(
      tmp, nullptr, nullptr, W_out, b_out, out, rowTiles);
}